// SECOND_BEAM_SEARCH_5909874999393
// MI455X (gfx1250) — compile-verified
//
#include <hip/hip_runtime.h>
#include <stdint.h>

// Problem constants (fixed by the reference).
#define L_      32
#define BEAM_   8
#define KV2_    2
#define HEADS_  8
#define SEQ_    1024
#define HDIM_   64
#define VOCAB_  50257
#define HIST_   128
#define TOPK_   8

// Derived.
#define INNER_FLOATS   (KV2_*HEADS_*SEQ_*HDIM_)      // 1,048,576 floats per (l,b)
#define INNER_VEC4     (INNER_FLOATS/4)              // 262,144 vec4 per (l,b)
#define N_KV           ((long long)L_*BEAM_*INNER_FLOATS) // 268,435,456 floats
#define TOTAL_VEC4     (N_KV/4)                      // 67,108,864
#define BLOCK_VEC4     16384                         // vec4 per block (256 KB)
#define GATHER_BLOCKS  (TOTAL_VEC4/BLOCK_VEC4)       // 4096
#define KDEPTH         4                             // async batch depth per thread
#define BATCH_VEC4     (256*KDEPTH)                  // 1024 vec4 per batch
#define NBATCH         (BLOCK_VEC4/BATCH_VEC4)       // 16 batches per block

// ---------------------------------------------------------------------------
// CDNA5 async LDS DMA helpers (GLOBAL_LOAD_ASYNC_TO_LDS_B128 /
// GLOBAL_STORE_ASYNC_FROM_LDS_B128, tracked with ASYNCcnt). These bypass
// VGPRs entirely: global -> LDS -> global with only addresses in the wave.
// ---------------------------------------------------------------------------
__device__ __forceinline__ void async_load_b128(uint32_t lds_off, const void* gaddr) {
    asm volatile("global_load_async_to_lds_b128 %0, %1, off"
                 :: "v"(lds_off), "v"(gaddr) : "memory");
}
__device__ __forceinline__ void async_store_b128(void* gaddr, uint32_t lds_off) {
    asm volatile("global_store_async_from_lds_b128 %0, %1, off"
                 :: "v"(gaddr), "v"(lds_off) : "memory");
}
__device__ __forceinline__ void wait_async0() {
    asm volatile("s_wait_asynccnt 0" ::: "memory");
}

// ---------------------------------------------------------------------------
// Kernel 1: per-row logsumexp + top-8 (8 blocks, one per beam row).
// Iterative block-argmax with an order-key exclusion reproduces
// jax.lax.top_k (descending values, lower index wins ties).
// Stores current_prob = logit - lse + previous_prob and the token index.
// ---------------------------------------------------------------------------
__global__ void topk_lse_kernel(const float* __restrict__ logits,
                                const float* __restrict__ prev_prob,
                                float* __restrict__ ws_prob,
                                int*   __restrict__ ws_idx) {
    const int row = blockIdx.x;
    const int tid = threadIdx.x;
    const float* x = logits + (long long)row * VOCAB_;

    __shared__ float sv[256];
    __shared__ int   si[256];

    // Pass 1: row max
    float m = -__builtin_inff();
    for (int i = tid; i < VOCAB_; i += 256) m = fmaxf(m, x[i]);
    sv[tid] = m; __syncthreads();
    for (int s = 128; s > 0; s >>= 1) {
        if (tid < s) sv[tid] = fmaxf(sv[tid], sv[tid + s]);
        __syncthreads();
    }
    m = sv[0]; __syncthreads();

    // Pass 2: sum of exp(x - m)
    float sum = 0.f;
    for (int i = tid; i < VOCAB_; i += 256) sum += expf(x[i] - m);
    sv[tid] = sum; __syncthreads();
    for (int s = 128; s > 0; s >>= 1) {
        if (tid < s) sv[tid] += sv[tid + s];
        __syncthreads();
    }
    const float lse = m + logf(sv[0]);
    const float pp  = prev_prob[row];
    __syncthreads();

    // Pass 3: iterative top-8 via order-key exclusion.
    float prev_v = __builtin_inff();
    int   prev_i = -1;
    for (int t = 0; t < TOPK_; ++t) {
        float bv = -__builtin_inff();
        int   bi = VOCAB_;
        for (int i = tid; i < VOCAB_; i += 256) {
            float v = x[i];
            bool ok = (v < prev_v) || (v == prev_v && i > prev_i);
            if (ok && (v > bv || (v == bv && i < bi))) { bv = v; bi = i; }
        }
        sv[tid] = bv; si[tid] = bi; __syncthreads();
        for (int s = 128; s > 0; s >>= 1) {
            if (tid < s) {
                if (sv[tid + s] > sv[tid] ||
                    (sv[tid + s] == sv[tid] && si[tid + s] < si[tid])) {
                    sv[tid] = sv[tid + s]; si[tid] = si[tid + s];
                }
            }
            __syncthreads();
        }
        prev_v = sv[0]; prev_i = si[0];
        if (tid == 0) {
            ws_prob[row * TOPK_ + t] = prev_v - lse + pp;   // current_prob
            ws_idx [row * TOPK_ + t] = prev_i;
        }
        __syncthreads();
    }
}

// ---------------------------------------------------------------------------
// Kernel 2: top-8 of the 64 candidates; emit top_beam_prob, tbi,
// max_logits_idx, beam_index (to ws) and new_save_id (8 x 129).
// ---------------------------------------------------------------------------
__global__ void beam_select_kernel(const float* __restrict__ ws_prob,
                                   const int*   __restrict__ ws_idx,
                                   const int*   __restrict__ save_id,
                                   int*   __restrict__ ws_beam_index,
                                   int*   __restrict__ sid_out,
                                   float* __restrict__ prob_out,
                                   int*   __restrict__ tbi_out,
                                   int*   __restrict__ mli_out) {
    __shared__ float cp[BEAM_ * TOPK_];
    __shared__ int   bi[BEAM_];
    __shared__ int   tbi[BEAM_];
    const int tid = threadIdx.x;   // 64 threads

    if (tid < BEAM_ * TOPK_) cp[tid] = ws_prob[tid];
    __syncthreads();

    if (tid == 0) {
        float pv = __builtin_inff();
        int   pi = -1;
        for (int t = 0; t < BEAM_; ++t) {
            float bv = -__builtin_inff();
            int   bj = BEAM_ * TOPK_;
            for (int j = 0; j < BEAM_ * TOPK_; ++j) {
                float v = cp[j];
                bool ok = (v < pv) || (v == pv && j > pi);
                if (ok && (v > bv || (v == bv && j < bj))) { bv = v; bj = j; }
            }
            pv = bv; pi = bj;
            bi[t]  = bj / TOPK_;
            tbi[t] = ws_idx[bj];
            prob_out[t]      = bv;
            tbi_out[t]       = tbi[t];
            ws_beam_index[t] = bi[t];
        }
        mli_out[0] = tbi[0];
    }
    __syncthreads();

    // new_save_id = concat(save_id[beam_index], tbi)  ->  8 rows x 129
    for (int j = 0; j < BEAM_; ++j) {
        for (int c = tid; c < HIST_; c += 64)
            sid_out[j * (HIST_ + 1) + c] = save_id[bi[j] * HIST_ + c];
    }
    if (tid < BEAM_) sid_out[tid * (HIST_ + 1) + HIST_] = tbi[tid];
}

// ---------------------------------------------------------------------------
// Kernel 3: 1 GiB kv_cache beam gather through the CDNA5 async-LDS DMA path,
// double-buffered so stores of batch i overlap loads of batch i+1 in flight
// on ASYNCcnt. The `s_wait_asynccnt 0` at the loop top both confirms batch
// i's loads landed in LDS and that the buffer being refilled has drained its
// previous stores (required: loads vs stores complete out of order).
// Only the beam coordinate differs between dst and src:
//   src_vec4 = dst_vec4 + (beam_index[b] - b) * INNER_VEC4
// ---------------------------------------------------------------------------
__global__ void kv_gather_kernel(const float* __restrict__ kv,
                                 float*       __restrict__ out,
                                 const int*   __restrict__ beam_index) {
    __shared__ __align__(16) float4 stage[2][256 * KDEPTH];   // 2 x 16 KB staging
    const int tid = threadIdx.x;

    const long long base  = (long long)blockIdx.x * BLOCK_VEC4;
    const int b_out       = (int)((base / INNER_VEC4) & (BEAM_ - 1));
    const long long delta = ((long long)beam_index[b_out] - b_out) * INNER_VEC4;

    const float4* src = (const float4*)kv + delta;
    float4*       dst = (float4*)out;

    // Prologue: loads for batch 0 into buffer 0.
#pragma unroll
    for (int k = 0; k < KDEPTH; ++k) {
        const long long v = base + (long long)k * 256 + tid;
        const uint32_t lds = (uint32_t)(uintptr_t)&stage[0][k * 256 + tid];
        async_load_b128(lds, (const void*)(src + v));
    }

    for (int i = 0; i < NBATCH; ++i) {
        wait_async0();                   // batch i in LDS; other buffer drained
        const int cur = i & 1;
        const long long bcur = base + (long long)i * BATCH_VEC4;
#pragma unroll
        for (int k = 0; k < KDEPTH; ++k) {
            const long long v = bcur + (long long)k * 256 + tid;
            const uint32_t lds = (uint32_t)(uintptr_t)&stage[cur][k * 256 + tid];
            async_store_b128((void*)(dst + v), lds);
        }
        if (i + 1 < NBATCH) {
            const long long bnext = bcur + BATCH_VEC4;
#pragma unroll
            for (int k = 0; k < KDEPTH; ++k) {
                const long long v = bnext + (long long)k * 256 + tid;
                const uint32_t lds = (uint32_t)(uintptr_t)&stage[cur ^ 1][k * 256 + tid];
                async_load_b128(lds, (const void*)(src + v));
            }
        }
    }
    wait_async0();                       // final stores complete
}

// ---------------------------------------------------------------------------
extern "C" void kernel_launch(void* const* d_in, const int* in_sizes, int n_in,
                              void* d_out, int out_size, void* d_ws, size_t ws_size,
                              hipStream_t stream) {
    const float* kv       = (const float*)d_in[0];
    const float* logits   = (const float*)d_in[1];
    const int*   save_id  = (const int*)d_in[2];
    const float* prev     = (const float*)d_in[3];
    // d_in[4]=beam_size(8), d_in[5]=top_k(8): fixed constants, baked in.

    // Output layout (flat, return order): kv_gathered | new_save_id |
    // top_beam_prob | tbi | max_logits_idx
    float* out_f   = (float*)d_out;
    float* kv_out  = out_f;
    int*   sid_out = (int*)(out_f + N_KV);
    float* prob_out = (float*)(sid_out + BEAM_ * (HIST_ + 1));
    int*   tbi_out = (int*)(prob_out + BEAM_);
    int*   mli_out = tbi_out + BEAM_;

    // Workspace layout.
    float* ws_prob = (float*)d_ws;                       // 64 floats
    int*   ws_idx  = (int*)((char*)d_ws + 256);          // 64 ints
    int*   ws_bi   = (int*)((char*)d_ws + 512);          // 8 ints

    topk_lse_kernel<<<BEAM_, 256, 0, stream>>>(logits, prev, ws_prob, ws_idx);
    beam_select_kernel<<<1, 64, 0, stream>>>(ws_prob, ws_idx, save_id,
                                             ws_bi, sid_out, prob_out, tbi_out, mli_out);
    kv_gather_kernel<<<GATHER_BLOCKS, 256, 0, stream>>>(kv, kv_out, ws_bi);
}